// COMET_14199161880837
// MI455X (gfx1250) — compile-verified
//
#include <hip/hip_runtime.h>
#include <hip/hip_bf16.h>

typedef __attribute__((ext_vector_type(2))) float v2f;
typedef __attribute__((ext_vector_type(8))) float v8f;

#define D_IN   1024
#define D_OUT  768
#define NB_E   16
#define NB_B   8192
#define NTILES 8           // 4 sel tiles (60 cols + 4 pad) + 4 leaf tiles (64 cols)
#define PART_STRIDE 40     // 16 soft + 16 hard + 1 reg, padded

__device__ __forceinline__ float smooth_step_f(float t) {
    // gamma = 1:  -2 t^3 + 1.5 t + 0.5, clamped at +-0.5
    float poly = fmaf(-2.0f * t * t, t, fmaf(1.5f, t, 0.5f));
    return (t <= -0.5f) ? 0.0f : ((t >= 0.5f) ? 1.0f : poly);
}

// ---------------------------------------------------------------------------
// Pack [sel_w | pad | leaf_w] into WMMA B-fragment friendly layout:
// Wt[tile][ (k>>1)*32 + c*2 + (k&1) ]  == W[col = tile*16+c][k]
// so a lane's (b0,b1) pair (K=k+2g, k+2g+1) is one contiguous float2.
// ---------------------------------------------------------------------------
__global__ __launch_bounds__(256) void COMET_prep(const float* __restrict__ sel_w,
                                                  const float* __restrict__ leaf_w,
                                                  float* __restrict__ Wt) {
    int idx = blockIdx.x * 256 + threadIdx.x;      // 8 * 16384 = 131072 total
    int tile   = idx >> 14;
    int rem    = idx & 16383;
    int pair   = rem >> 5;
    int within = rem & 31;
    int c = within >> 1;
    int k = pair * 2 + (within & 1);
    float v = 0.0f;
    if (tile < 4) {
        int col = tile * 16 + c;                   // n*4 + kk, n<15 -> col<60
        if (col < 60) v = sel_w[(size_t)col * D_IN + k];
    } else {
        int col = (tile - 4) * 16 + c;             // e*4 + kk
        v = leaf_w[(size_t)col * D_IN + k];
    }
    Wt[idx] = v;
}

// ---------------------------------------------------------------------------
// WMMA gating kernel: 16 rows per block, 4 waves x 2 n-tiles each.
// Scores tile (16 x 128) -> LDS -> per-row tree / softmax on lanes 0..15.
// ---------------------------------------------------------------------------
__global__ __launch_bounds__(128) void COMET_gate(const float* __restrict__ x,
                                                  const float* __restrict__ Wt,
                                                  const float* __restrict__ leaf_b,
                                                  float* __restrict__ wnorm,
                                                  float* __restrict__ partial,
                                                  float* __restrict__ s_concat) {
    __shared__ float S[16][128];
    __shared__ float WN[16][16];
    __shared__ float RREG[16];

    const int tid  = threadIdx.x;
    const int lane = tid & 31;
    const int wv   = tid >> 5;
    const int g    = lane >> 4;     // k-group 0/1
    const int c    = lane & 15;     // M (for A) / N (for B)
    const int row0 = blockIdx.x * 16;

    // A fragment source: x[row0+c][k + 2g], contiguous float2 per step
    const float* abase  = x + (size_t)(row0 + c) * D_IN + 2 * g;
    const int t0 = 2 * wv, t1 = 2 * wv + 1;
    const float* bbase0 = Wt + (size_t)t0 * (D_IN * 16) + g * 32 + c * 2;
    const float* bbase1 = Wt + (size_t)t1 * (D_IN * 16) + g * 32 + c * 2;

    v8f acc0 = {};
    v8f acc1 = {};
#pragma unroll 8
    for (int k = 0; k < D_IN; k += 4) {
        v2f a  = *(const v2f*)(abase + k);
        v2f b0 = *(const v2f*)(bbase0 + (size_t)k * 16);
        v2f b1 = *(const v2f*)(bbase1 + (size_t)k * 16);
        acc0 = __builtin_amdgcn_wmma_f32_16x16x4_f32(false, a, false, b0,
                                                     (short)0, acc0, false, false);
        acc1 = __builtin_amdgcn_wmma_f32_16x16x4_f32(false, a, false, b1,
                                                     (short)0, acc1, false, false);
    }

    // D layout: VGPR i -> row (i + 8g), col = lane&15 within tile
#pragma unroll
    for (int i = 0; i < 8; i++) {
        int m = i + g * 8;
        S[m][t0 * 16 + c] = acc0[i];
        S[m][t1 * 16 + c] = acc1[i];
    }
    __syncthreads();

    if (tid < 16) {
        const int r = tid;
        const size_t brow = (size_t)(row0 + r);
        float f[64];
        // -------- tree probabilities per k-channel --------
#pragma unroll
        for (int kk = 0; kk < 4; kk++) {
            float pr[16];
            pr[0] = 1.0f;
            int start = 0;
#pragma unroll
            for (int d = 0; d < 4; d++) {
                int n = 1 << d;
                for (int j = n - 1; j >= 0; j--) {
                    float gg = smooth_step_f(S[r][(start + j) * 4 + kk]);
                    float pj = pr[j];
                    pr[2 * j]     = pj * gg;
                    pr[2 * j + 1] = pj * (1.0f - gg);
                }
                start += n;
            }
#pragma unroll
            for (int e = 0; e < 16; e++) f[e * 4 + kk] = pr[e];
        }
        // -------- entropy reg + scores s = a + logp --------
        float rreg = 0.0f;
#pragma unroll
        for (int i = 0; i < 64; i++) {
            float p  = f[i];
            float pc = fmaxf(p, 1e-6f) + 1e-6f;     // clip(p, EPS) + EPS
            rreg -= pc * __logf(pc);
            float logp = (p < 1e-5f) ? -__builtin_inff() : __logf(p + 1e-6f);
            f[i] = S[r][64 + i] + leaf_b[i] + logp;
        }
        // -------- softmax over 64 --------
        float m = -__builtin_inff();
#pragma unroll
        for (int i = 0; i < 64; i++) m = fmaxf(m, f[i]);
        float sum = 0.0f;
#pragma unroll
        for (int i = 0; i < 64; i++) { float w = __expf(f[i] - m); f[i] = w; sum += w; }
        float inv = 1.0f / sum;
        float we[16];
        float tot = 0.0f;
#pragma unroll
        for (int e = 0; e < 16; e++) {
            float s4 = (f[4 * e] + f[4 * e + 1]) + (f[4 * e + 2] + f[4 * e + 3]);
            we[e] = s4 * inv;
            tot += we[e];
        }
        float invt = 1.0f / tot;
#pragma unroll
        for (int e = 0; e < 16; e++) {
            float wn = we[e] * invt;
            wnorm[brow * 16 + e]    = wn;
            s_concat[brow * 16 + e] = (wn < 1e-5f) ? 1.0f : 0.0f;
            WN[r][e] = wn;
        }
        RREG[r] = rreg;
    }
    __syncthreads();

    // per-block partials (fixed-order sums -> deterministic)
    if (tid < 16) {
        float ss = 0.0f, hh = 0.0f;
#pragma unroll
        for (int r = 0; r < 16; r++) {
            float w = WN[r][tid];
            ss += w;
            hh += (w < 1e-5f) ? 0.0f : 1.0f;
        }
        partial[blockIdx.x * PART_STRIDE + tid]      = ss;
        partial[blockIdx.x * PART_STRIDE + 16 + tid] = hh;
    } else if (tid == 32) {
        float rr = 0.0f;
#pragma unroll
        for (int r = 0; r < 16; r++) rr += RREG[r];
        partial[blockIdx.x * PART_STRIDE + 32] = rr;
    }
}

// ---------------------------------------------------------------------------
// y_agg[b,d] = sum_e h[b,d,e] * wnorm[b,e]   -- the 402 MB streaming kernel.
// grid (3, 8192), 256 threads; each thread: 4x float4 contiguous loads.
// ---------------------------------------------------------------------------
__global__ __launch_bounds__(256) void COMET_yagg(const float* __restrict__ h,
                                                  const float* __restrict__ wnorm,
                                                  float* __restrict__ y) {
    __shared__ float wn[16];
    const int b = blockIdx.y;
    if (threadIdx.x < 16) wn[threadIdx.x] = wnorm[(size_t)b * 16 + threadIdx.x];
    __syncthreads();
    const int d = blockIdx.x * 256 + threadIdx.x;          // 3*256 = 768
    const size_t bd = (size_t)b * D_OUT + d;
    const float4* hp = (const float4*)(h + (bd << 4));
    float4 h0 = hp[0], h1 = hp[1], h2 = hp[2], h3 = hp[3];
    float acc = h0.x * wn[0]  + h0.y * wn[1]  + h0.z * wn[2]  + h0.w * wn[3]
              + h1.x * wn[4]  + h1.y * wn[5]  + h1.z * wn[6]  + h1.w * wn[7]
              + h2.x * wn[8]  + h2.y * wn[9]  + h2.z * wn[10] + h2.w * wn[11]
              + h3.x * wn[12] + h3.y * wn[13] + h3.z * wn[14] + h3.w * wn[15];
    y[bd] = acc;
}

// ---------------------------------------------------------------------------
// Final reduction of 512 block partials (fixed order -> deterministic).
// ---------------------------------------------------------------------------
__global__ __launch_bounds__(64) void COMET_reduce(const float* __restrict__ partial,
                                                   float* __restrict__ soft,
                                                   float* __restrict__ hard,
                                                   float* __restrict__ reg) {
    int t = threadIdx.x;
    if (t >= 33) return;
    float s = 0.0f;
    for (int i = 0; i < 512; i++) s += partial[i * PART_STRIDE + t];
    const float invB = 1.0f / (float)NB_B;
    if (t < 16)       soft[t] = s * invB;
    else if (t < 32)  hard[t - 16] = s * invB;
    else              *reg = s * (0.01f * invB);     // ENTROPY_REG * mean_b
}

extern "C" void kernel_launch(void* const* d_in, const int* in_sizes, int n_in,
                              void* d_out, int out_size, void* d_ws, size_t ws_size,
                              hipStream_t stream) {
    const float* h      = (const float*)d_in[0];   // (8192, 768, 16)
    const float* x      = (const float*)d_in[1];   // (8192, 1024)
    const float* sel_w  = (const float*)d_in[2];   // (15, 4, 1024)
    const float* leaf_w = (const float*)d_in[3];   // (16, 4, 1024)
    const float* leaf_b = (const float*)d_in[4];   // (16, 4)
    float* out = (float*)d_out;

    // output layout (flat concat, return order)
    float* y        = out;                               // 8192*768
    float* soft     = out + (size_t)NB_B * D_OUT;        // 16
    float* hard     = soft + 16;                         // 16
    float* s_concat = hard + 16;                         // 8192*16
    float* reg      = s_concat + (size_t)NB_B * NB_E;    // 1

    // workspace layout
    float* Wt      = (float*)d_ws;                       // 8*1024*16 = 131072
    float* wnorm   = Wt + NTILES * D_IN * 16;            // 8192*16   = 131072
    float* partial = wnorm + (size_t)NB_B * NB_E;        // 512*40    = 20480

    COMET_prep  <<<512, 256, 0, stream>>>(sel_w, leaf_w, Wt);
    COMET_gate  <<<NB_B / 16, 128, 0, stream>>>(x, Wt, leaf_b, wnorm, partial, s_concat);
    COMET_yagg  <<<dim3(D_OUT / 256, NB_B), 256, 0, stream>>>(h, wnorm, y);
    COMET_reduce<<<1, 64, 0, stream>>>(partial, soft, hard, reg);
}